// Renderer_48644799594646
// MI455X (gfx1250) — compile-verified
//
#include <hip/hip_runtime.h>
#include <math.h>

// ---------------------------------------------------------------------------
// Renderer for MI455X (gfx1250). Bandwidth-bound (~270MB traffic -> ~11us
// floor @ 23.3 TB/s). Fused per-tile kernel keeps wc1 in LDS; the 16x16
// weighted reduction uses full-precision V_WMMA_F32_16X16X4_F32.
// ---------------------------------------------------------------------------

typedef __attribute__((ext_vector_type(2))) float v2f;
typedef __attribute__((ext_vector_type(8))) float v8f;

#define NPIX   (128 * 128)

// ---- 3x3 symmetric eigen plane-fit (smallest eigenvector, z >= 0) ---------
__device__ __forceinline__ void plane_fit9(const float* xs, const float* ys,
                                           const float* zs, float out[3]) {
    double mx = 0.0, my = 0.0, mz = 0.0;
#pragma unroll
    for (int k = 0; k < 9; ++k) { mx += xs[k]; my += ys[k]; mz += zs[k]; }
    mx *= (1.0 / 9.0); my *= (1.0 / 9.0); mz *= (1.0 / 9.0);

    double cxx = 0, cxy = 0, cxz = 0, cyy = 0, cyz = 0, czz = 0;
#pragma unroll
    for (int k = 0; k < 9; ++k) {
        double dx = (double)xs[k] - mx;
        double dy = (double)ys[k] - my;
        double dz = (double)zs[k] - mz;
        cxx += dx * dx; cxy += dx * dy; cxz += dx * dz;
        cyy += dy * dy; cyz += dy * dz; czz += dz * dz;
    }

    double vx = 0.0, vy = 0.0, vz = 1.0;
    double p1 = cxy * cxy + cxz * cxz + cyz * cyz;
    double q  = (cxx + cyy + czz) * (1.0 / 3.0);
    double p2 = (cxx - q) * (cxx - q) + (cyy - q) * (cyy - q) +
                (czz - q) * (czz - q) + 2.0 * p1;
    if (p2 > 1e-300) {
        double p  = sqrt(p2 * (1.0 / 6.0));
        double ip = 1.0 / p;
        double b00 = (cxx - q) * ip, b11 = (cyy - q) * ip, b22 = (czz - q) * ip;
        double b01 = cxy * ip, b02 = cxz * ip, b12 = cyz * ip;
        double det = b00 * (b11 * b22 - b12 * b12)
                   - b01 * (b01 * b22 - b12 * b02)
                   + b02 * (b01 * b12 - b11 * b02);
        double r = 0.5 * det;
        r = fmin(1.0, fmax(-1.0, r));
        double ang  = acos(r) * (1.0 / 3.0);
        double lmin = q + 2.0 * p * cos(ang + 2.0943951023931953); // +2pi/3
        // M = Cov - lmin*I ; eigenvector = best cross product of two rows
        double m00 = cxx - lmin, m11 = cyy - lmin, m22 = czz - lmin;
        double c0x = cxy * cyz - cxz * m11;
        double c0y = cxz * cxy - m00 * cyz;
        double c0z = m00 * m11 - cxy * cxy;
        double c1x = m11 * m22 - cyz * cyz;
        double c1y = cyz * cxz - cxy * m22;
        double c1z = cxy * cyz - m11 * cxz;
        double c2x = cyz * cxy - cxz * m11;   // r2 x r0 components
        double c2y = cxz * cxz - m22 * m00;
        double c2z = m22 * cxy - cxz * cyz;
        // recompute c2 correctly: r2=(cxz,cyz,m22), r0=(m00,cxy,cxz)
        c2x = cyz * cxz - m22 * cxy;
        c2y = m22 * m00 - cxz * cxz;
        c2z = cxz * cxy - cyz * m00;
        double n0s = c0x * c0x + c0y * c0y + c0z * c0z;
        double n1s = c1x * c1x + c1y * c1y + c1z * c1z;
        double n2s = c2x * c2x + c2y * c2y + c2z * c2z;
        double bx = c0x, by = c0y, bz = c0z, bn = n0s;
        if (n1s > bn) { bx = c1x; by = c1y; bz = c1z; bn = n1s; }
        if (n2s > bn) { bx = c2x; by = c2y; bz = c2z; bn = n2s; }
        if (bn > 1e-300) {
            double inv = 1.0 / sqrt(bn);
            vx = bx * inv; vy = by * inv; vz = bz * inv;
        }
    } else {
        vx = 1.0; vy = 0.0; vz = 0.0;   // Cov ~ q*I : eigh returns e0=(1,0,0)
    }
    if (vz < 0.0) { vx = -vx; vy = -vy; vz = -vz; }
    out[0] = (float)vx; out[1] = (float)vy; out[2] = (float)vz;
}

// ---- Kernel 1: wc0 = unproject depth0 through perspective rays ------------
__global__ void k_wc0(const float* __restrict__ view,
                      const float* __restrict__ depth0,
                      float* __restrict__ wc0) {
    int idx = blockIdx.x * blockDim.x + threadIdx.x;
    if (idx >= NPIX) return;
    int i = idx >> 7, j = idx & 127;
    const float scale = 0.026794919243112274f;   // hh/2 = ww/2 (fov 30, F=0.1)
    float x = (-1.0f + 2.0f * (float)i * (1.0f / 127.0f)) * scale;
    float y = ( 1.0f - 2.0f * (float)j * (1.0f / 127.0f)) * scale;
    float phi = atan2f(x, y);
    const float st = 0.9949874371066199f;        // sin(acos(0.1)) = sqrt(0.99)
    float dx = view[3] + st * cosf(phi);
    float dy = view[4] + st * sinf(phi);
    float dz = view[5];
    float inv = 1.0f / sqrtf(dx * dx + dy * dy + dz * dz);
    float dep = depth0[idx];
    wc0[idx * 3 + 0] = view[0] + dep * dx * inv;
    wc0[idx * 3 + 1] = view[1] + dep * dy * inv;
    wc0[idx * 3 + 2] = view[2] + dep * dz * inv;
}

// ---- Kernel 2: n0 = plane-fit normals on the 128x128 wc0 grid -------------
__global__ void k_n0(const float* __restrict__ wc0, float* __restrict__ n0v) {
    int idx = blockIdx.x * blockDim.x + threadIdx.x;
    if (idx >= NPIX) return;
    int i = idx >> 7, j = idx & 127;
    float xs[9], ys[9], zs[9];
    int k = 0;
#pragma unroll
    for (int di = -1; di <= 1; ++di) {
#pragma unroll
        for (int dj = -1; dj <= 1; ++dj) {
            int ii = i + di; ii = ii < 0 ? 0 : (ii > 127 ? 127 : ii);
            int jj = j + dj; jj = jj < 0 ? 0 : (jj > 127 ? 127 : jj);
            const float* p = wc0 + (ii * 128 + jj) * 3;
            xs[k] = p[0]; ys[k] = p[1]; zs[k] = p[2]; ++k;
        }
    }
    float n[3];
    plane_fit9(xs, ys, zs, n);
    float* o = n0v + idx * 3;
    o[0] = n[0]; o[1] = n[1]; o[2] = n[2];
}

// ---- Kernel 3: fused wc1 -> (n1, out0) per 16x16 tile ---------------------
__global__ void __launch_bounds__(256)
k_main(const float* __restrict__ depth1, const float* __restrict__ irr,
       const float* __restrict__ wc0g,  const float* __restrict__ n0g,
       float* __restrict__ out0,        float* __restrict__ n1out) {
    __shared__ float s_wc[3][256];
    __shared__ float s_p[3][256];

    int pix = blockIdx.x;
    int t = threadIdx.x;
    int u = t >> 4, v = t & 15;

    const float* w0  = wc0g + pix * 3;
    const float* nn0 = n0g  + pix * 3;
    float cx = w0[0],  cy = w0[1],  cz = w0[2];
    float nx = nn0[0], ny = nn0[1], nz = nn0[2];

    // fisheye_angles(16,16): phi = u*2pi/15, theta = v*pi/15
    float pu = (float)u * 0.4188790204786391f;
    float tv = (float)v * 0.20943951023931953f;
    float st = sinf(tv);
    float dx = nx + st * cosf(pu);
    float dy = ny + st * sinf(pu);
    float dz = nz;
    float inv = 1.0f / sqrtf(dx * dx + dy * dy + dz * dz);
    float dep = depth1[(size_t)pix * 256 + t];
    s_wc[0][t] = cx + dep * dx * inv;
    s_wc[1][t] = cy + dep * dy * inv;
    s_wc[2][t] = cz + dep * dz * inv;

    // irradiance partial: theta2 = {0, pi}; weight(b=0) is exactly 0 -> skip.
    const float PIF = 3.14159265358979323846f;       // rounds to f32 pi
    float w2  = cosf(PIF) * sinf(PIF);               // tiny but nonzero
    float w1v = cosf(tv) * sinf(tv);
    const float* ir = irr + ((size_t)pix * 256 + t) * 12;
#pragma unroll
    for (int c = 0; c < 3; ++c) {
        float s  = w2 * (ir[3 + c] + ir[9 + c]);     // only b==1 terms survive
        float o1 = 0.1f + 0.125f * s;                // emi + alb/(W2*H2)*sum
        s_p[c][t] = o1 * w1v;
    }
    __syncthreads();

    // ---- wave 0: out0 via full-precision V_WMMA_F32_16X16X4_F32 ----------
    // D = sum_k A(16x16 tile, in 4 K-steps) * ones(4x16); every column of D
    // is sum_v A[m][v]; fold 8 C/D VGPRs + xor-16 shuffle to get sum_{m,v}.
    if (t < 32) {
        int lane = t;
        int m  = lane & 15;
        int kh = lane >> 4;       // A layout: lanes 0-15 hold K {0,1}, 16-31 {2,3}
        float tot[3];
#pragma unroll
        for (int c = 0; c < 3; ++c) {
            v8f acc = {0.f, 0.f, 0.f, 0.f, 0.f, 0.f, 0.f, 0.f};
#pragma unroll
            for (int kk = 0; kk < 4; ++kk) {
                v2f a;
                a[0] = s_p[c][m * 16 + kk * 4 + kh * 2 + 0];
                a[1] = s_p[c][m * 16 + kk * 4 + kh * 2 + 1];
                v2f b; b[0] = 1.0f; b[1] = 1.0f;
                acc = __builtin_amdgcn_wmma_f32_16x16x4_f32(
                    false, a, false, b, (short)0, acc, false, false);
            }
            float s = acc[0] + acc[1] + acc[2] + acc[3] +
                      acc[4] + acc[5] + acc[6] + acc[7];
            s += __shfl_xor(s, 16, 32);              // join M halves (wave32)
            tot[c] = s;
        }
        if (lane == 0) {
#pragma unroll
            for (int c = 0; c < 3; ++c)
                out0[pix * 3 + c] = 0.1f + (0.5f / 256.0f) * tot[c];
        }
    }

    // ---- all threads: n1 plane fit on the LDS wc1 tile (edge clamp) ------
    float xs[9], ys[9], zs[9];
    int k = 0;
#pragma unroll
    for (int du = -1; du <= 1; ++du) {
#pragma unroll
        for (int dv = -1; dv <= 1; ++dv) {
            int uu = u + du; uu = uu < 0 ? 0 : (uu > 15 ? 15 : uu);
            int vv = v + dv; vv = vv < 0 ? 0 : (vv > 15 ? 15 : vv);
            int tt = uu * 16 + vv;
            xs[k] = s_wc[0][tt]; ys[k] = s_wc[1][tt]; zs[k] = s_wc[2][tt]; ++k;
        }
    }
    float n[3];
    plane_fit9(xs, ys, zs, n);
    float* o = n1out + ((size_t)pix * 256 + t) * 3;
    o[0] = n[0]; o[1] = n[1]; o[2] = n[2];
}

// ---------------------------------------------------------------------------
extern "C" void kernel_launch(void* const* d_in, const int* in_sizes, int n_in,
                              void* d_out, int out_size, void* d_ws,
                              size_t ws_size, hipStream_t stream) {
    (void)in_sizes; (void)n_in; (void)out_size; (void)ws_size;
    const float* view   = (const float*)d_in[0];
    const float* depth0 = (const float*)d_in[1];
    const float* depth1 = (const float*)d_in[2];
    const float* irr    = (const float*)d_in[3];
    float* out = (float*)d_out;            // [out0 : 128*128*3][n1 : ...*16*16*3]
    float* wc0 = (float*)d_ws;             // 128*128*3 floats
    float* n0  = wc0 + NPIX * 3;           // 128*128*3 floats

    k_wc0<<<NPIX / 256, 256, 0, stream>>>(view, depth0, wc0);
    k_n0 <<<NPIX / 256, 256, 0, stream>>>(wc0, n0);
    k_main<<<NPIX, 256, 0, stream>>>(depth1, irr, wc0, n0, out, out + NPIX * 3);
}